// EdgeEmbedding_7928509628538
// MI455X (gfx1250) — compile-verified
//
#include <hip/hip_runtime.h>
#include <hip/hip_bf16.h>
#include <math.h>

typedef __attribute__((ext_vector_type(2))) float v2f;
typedef __attribute__((ext_vector_type(8))) float v8f;

#define B_    32
#define N_    1024
#define K_    204            // max(N//5, 10)
#define ED_   32
#define RPW   16             // rows per workgroup
#define E_TOT (B_ * N_ * K_) // 6,684,672 edges
#define X_SZ  (B_ * N_ * ED_)// 1,048,576 node-feature floats

// One workgroup handles a 16-row strip of one batch's distance matrix.
// Phase A: WMMA f32 16x16x4 computes |xi-xj|^2 tiles into LDS.
// Phase B: exact rank-based top-k (jax top_k tie-break) + fused edge outputs.
__global__ __launch_bounds__(256) void dist_topk_edges_kernel(
    const float* __restrict__ locs,   // (B, N, 2)
    const float* __restrict__ W,      // (32, 1) flat
    const float* __restrict__ bias,   // (32,)
    float* __restrict__ out)          // full concatenated output
{
    __shared__ float sq[RPW][N_];     // 64 KB squared-distance strip
    __shared__ float Ws[ED_], bs[ED_];

    const int tid    = threadIdx.x;
    const int lane   = tid & 31;
    const int wave   = tid >> 5;
    const int bb     = blockIdx.x >> 6;        // blockIdx / (N/16)
    const int rowBlk = blockIdx.x & 63;
    const int I0     = rowBlk * RPW;

    if (tid < ED_) { Ws[tid] = W[tid]; bs[tid] = bias[tid]; }

    // ---------------- Phase A: squared distances via WMMA ----------------
    const int m  = lane & 15;                  // row/col within 16-tile
    const int hi = lane >> 4;                  // 0: K0/K1 half, 1: K2/K3 half
    const float* lb = locs + (size_t)bb * N_ * 2;

    // A operand (rows I0..I0+15): [-2x, -2y | |p|^2, 1]
    const float ax = lb[(I0 + m) * 2 + 0];
    const float ay = lb[(I0 + m) * 2 + 1];
    v2f A;
    A.x = hi ? (ax * ax + ay * ay) : (-2.0f * ax);
    A.y = hi ? 1.0f                : (-2.0f * ay);

    for (int t = 0; t < 8; ++t) {
        const int J0 = (wave * 8 + t) * 16;
        // B operand (cols J0..J0+15): [x', y' | 1, |p'|^2]
        const float bx = lb[(J0 + m) * 2 + 0];
        const float by = lb[(J0 + m) * 2 + 1];
        v2f Bv;
        Bv.x = hi ? 1.0f                : bx;
        Bv.y = hi ? (bx * bx + by * by) : by;

        v8f C = {};
        C = __builtin_amdgcn_wmma_f32_16x16x4_f32(
                /*neg_a=*/false, A, /*neg_b=*/false, Bv,
                /*c_mod=*/(short)0, C, /*reuse_a=*/false, /*reuse_b=*/false);

        const int jn = J0 + m;                 // node index of this column
        #pragma unroll
        for (int v = 0; v < 8; ++v) {
            const int r  = v + 8 * hi;         // D layout: M = v + 8*(lane/16)
            const int in = I0 + r;
            const float s = fmaxf(C[v], 0.0f); // clamp FP cancellation
            sq[r][jn] = (in == jn) ? __builtin_inff() : s;  // mask self-loop
        }
    }
    __syncthreads();

    // ---------------- Phase B: exact top-k + edge emission ----------------
    // rank(j) = #{j' : s' < s} + #{j' < j : s' == s}  (matches jax top_k order)
    const size_t xOff = (size_t)X_SZ;
    for (int rr = 0; rr < 2; ++rr) {
        const int r      = wave * 2 + rr;      // 8 waves x 2 rows = 16 rows
        const int iNode  = I0 + r;
        const int srcGid = bb * N_ + iNode;

        for (int j = lane; j < N_; j += 32) {
            const float s = sq[r][j];
            if (s == __builtin_inff()) continue;   // self edge, never selected
            int rank = 0;
            for (int jp = 0; jp < N_; ++jp) {      // wave-uniform LDS broadcast
                const float sp = sq[r][jp];
                rank += (sp < s) || (sp == s && jp < j);
            }
            if (rank < K_) {
                const size_t e = (size_t)srcGid * K_ + (size_t)rank;
                out[xOff + e]                   = (float)srcGid;          // src
                out[xOff + (size_t)E_TOT + e]   = (float)(bb * N_ + j);   // dst
                const float val = sqrtf(s);
                float* emb = out + xOff + 2 * (size_t)E_TOT + e * ED_;
                #pragma unroll
                for (int c = 0; c < ED_; ++c)
                    emb[c] = fmaf(val, Ws[c], bs[c]);
            }
        }
    }
}

// x = init_embeddings.reshape(B*N, D) -> first X_SZ floats of d_out
__global__ void copy_x_kernel(const float* __restrict__ src,
                              float* __restrict__ out, int n)
{
    const int i = blockIdx.x * blockDim.x + threadIdx.x;
    if (i < n) out[i] = src[i];
}

extern "C" void kernel_launch(void* const* d_in, const int* in_sizes, int n_in,
                              void* d_out, int out_size, void* d_ws, size_t ws_size,
                              hipStream_t stream)
{
    (void)in_sizes; (void)n_in; (void)out_size; (void)d_ws; (void)ws_size;
    const float* locs = (const float*)d_in[0];   // (32,1024,2)
    const float* x    = (const float*)d_in[1];   // (32,1024,32)
    const float* W    = (const float*)d_in[2];   // (32,1)
    const float* bias = (const float*)d_in[3];   // (32,)
    float* out = (float*)d_out;

    copy_x_kernel<<<(X_SZ + 255) / 256, 256, 0, stream>>>(x, out, X_SZ);
    dist_topk_edges_kernel<<<B_ * (N_ / RPW), 256, 0, stream>>>(locs, W, bias, out);
}